// QMIParticleQAEEncoder_33380485825089
// MI455X (gfx1250) — compile-verified
//
#include <hip/hip_runtime.h>

// MI455X (gfx1250) QAE encoder.
// Roofline: 34 MB I/O (~1.5us @ 23.3 TB/s) vs ~15-20 GFLOP fp32 -> compute bound.
// Batch-invariant depth-section of the 7-qubit block folded into a fixed 128x128
// real unitary applied with V_WMMA_F32_16X16X4_F32 (f32 WMMA, matches complex64 ref).

typedef float v2f __attribute__((ext_vector_type(2)));
typedef float v8f __attribute__((ext_vector_type(8)));

#define DEVINL static __device__ __forceinline__

#if defined(__has_builtin)
#  if __has_builtin(__builtin_amdgcn_wmma_f32_16x16x4_f32)
#    define HAVE_WMMA_F32X4 1
#  endif
#endif

// ---- x-column index tables (qubit k of each sub-circuit) ----
__constant__ static const int cA_PT[4]  = {5, 4, 35, 34};
__constant__ static const int cA_ETA[4] = {9, 8, 45, 44};
__constant__ static const int cA_PHI[4] = {13, 12, 55, 54};
__constant__ static const int cB_PT[4]  = {3, 33, 31, 2};
__constant__ static const int cB_ETA[4] = {7, 43, 41, 6};
__constant__ static const int cB_PHI[4] = {11, 53, 51, 10};
__constant__ static const int cC_PT[4]  = {28, 32, 15, 16};
__constant__ static const int cC_ETA[4] = {38, 42, 19, 20};
__constant__ static const int cC_PHI[4] = {48, 52, 23, 24};
__constant__ static const int cD_PT[7]  = {0, 14, 30, 26, 29, 27, 17};
__constant__ static const int cD_ETA[7] = {-1, 18, 40, 36, 39, 37, 21};
__constant__ static const int cD_PHI[7] = {1, 22, 50, 46, 49, 47, 25};

// ---- small complex 2x2 helpers (fuse Rx/Ry/Rz per qubit into one gate) ----
struct C2 { float r, i; };
DEVINL C2 cmul(C2 a, C2 b) { return { a.r*b.r - a.i*b.i, a.r*b.i + a.i*b.r }; }
DEVINL C2 cadd(C2 a, C2 b) { return { a.r + b.r, a.i + b.i }; }
DEVINL void mm2(const C2 a[4], const C2 b[4], C2 o[4]) {
  o[0] = cadd(cmul(a[0], b[0]), cmul(a[1], b[2]));
  o[1] = cadd(cmul(a[0], b[1]), cmul(a[1], b[3]));
  o[2] = cadd(cmul(a[2], b[0]), cmul(a[3], b[2]));
  o[3] = cadd(cmul(a[2], b[1]), cmul(a[3], b[3]));
}
DEVINL void rot_rx(float th, C2 m[4]) {
  float c = __cosf(0.5f*th), s = __sinf(0.5f*th);
  m[0] = {c,0.f}; m[1] = {0.f,-s}; m[2] = {0.f,-s}; m[3] = {c,0.f};
}
DEVINL void rot_ry(float th, C2 m[4]) {
  float c = __cosf(0.5f*th), s = __sinf(0.5f*th);
  m[0] = {c,0.f}; m[1] = {-s,0.f}; m[2] = {s,0.f}; m[3] = {c,0.f};
}
DEVINL void rot_rz(float th, C2 m[4]) {
  float c = __cosf(0.5f*th), s = __sinf(0.5f*th);
  m[0] = {c,-s}; m[1] = {0.f,0.f}; m[2] = {0.f,0.f}; m[3] = {c,s};
}

// apply 2x2 complex gate on bit `mask` of an LDS statevector row
DEVINL void apply2(float* re, float* im, int mask, int npairs, const C2 m[4]) {
  for (int p = 0; p < npairs; ++p) {
    int i0 = ((p & ~(mask - 1)) << 1) | (p & (mask - 1));
    int i1 = i0 | mask;
    float r0 = re[i0], q0 = im[i0], r1 = re[i1], q1 = im[i1];
    re[i0] = m[0].r*r0 - m[0].i*q0 + m[1].r*r1 - m[1].i*q1;
    im[i0] = m[0].r*q0 + m[0].i*r0 + m[1].r*q1 + m[1].i*r1;
    re[i1] = m[2].r*r0 - m[2].i*q0 + m[3].r*r1 - m[3].i*q1;
    im[i1] = m[2].r*q0 + m[2].i*r0 + m[3].r*q1 + m[3].i*r1;
  }
}

// CNOT: swap amplitudes with ctrl=1, tgt=0 <-> ctrl=1, tgt=1. nswap = n/4.
DEVINL void cnot_c(float* re, float* im, int cmask, int tmask, int nswap) {
  int lo = cmask < tmask ? cmask : tmask;
  int hi = cmask < tmask ? tmask : cmask;
  for (int p = 0; p < nswap; ++p) {
    int t = ((p & ~(lo - 1)) << 1) | (p & (lo - 1));
    t = ((t & ~(hi - 1)) << 1) | (t & (hi - 1));
    int i = t | cmask;
    int j = i | tmask;
    float tmp = re[i]; re[i] = re[j]; re[j] = tmp;
    tmp = im[i]; im[i] = im[j]; im[j] = tmp;
  }
}

DEVINL void cnot_r(float* re, int cmask, int tmask, int nswap) {
  int lo = cmask < tmask ? cmask : tmask;
  int hi = cmask < tmask ? tmask : cmask;
  for (int p = 0; p < nswap; ++p) {
    int t = ((p & ~(lo - 1)) << 1) | (p & (lo - 1));
    t = ((t & ~(hi - 1)) << 1) | (t & (hi - 1));
    int i = t | cmask;
    int j = i | tmask;
    float tmp = re[i]; re[i] = re[j]; re[j] = tmp;
  }
}

// ---- full 4-qubit sub-circuit on an LDS statevector row ----
DEVINL void run_block4(const float* __restrict__ xrow,
                       const int* pti, const int* etai, const int* phii,
                       const float* __restrict__ w,
                       float* re, float* im, float* z) {
  for (int i = 0; i < 16; ++i) { re[i] = 0.f; im[i] = 0.f; }
  re[0] = 1.f;
  C2 a[4], b[4], t[4], m[4];
#pragma unroll
  for (int k = 0; k < 4; ++k) {                     // Rx(eta) Ry(pt) Rz(phi)
    int mask = 1 << (3 - k);
    float eta = (etai[k] >= 0) ? xrow[etai[k]] : 0.f;
    rot_rx(eta, a); rot_ry(xrow[pti[k]], b); mm2(b, a, t);
    rot_rz(xrow[phii[k]], a); mm2(a, t, m);
    apply2(re, im, mask, 8, m);
  }
#pragma unroll
  for (int k = 0; k < 3; ++k)                       // CNOT chain k -> k+1
    cnot_c(re, im, 1 << (3 - k), 1 << (2 - k), 4);
#pragma unroll
  for (int k = 0; k < 4; ++k) {                     // Rz(eta) Ry(pt) Rx(phi)
    int mask = 1 << (3 - k);
    float eta = (etai[k] >= 0) ? xrow[etai[k]] : 0.f;
    rot_rz(eta, a); rot_ry(xrow[pti[k]], b); mm2(b, a, t);
    rot_rx(xrow[phii[k]], a); mm2(a, t, m);
    apply2(re, im, mask, 8, m);
  }
  // depth-1: CNOT(t,l), Ry(w), CNOT(l,t)   trash={2,3}, latent={0,1}
#pragma unroll
  for (int tq = 2; tq <= 3; ++tq)
#pragma unroll
    for (int lq = 0; lq <= 1; ++lq)
      cnot_c(re, im, 1 << (3 - tq), 1 << (3 - lq), 4);
#pragma unroll
  for (int k = 0; k < 4; ++k) {
    rot_ry(w[k], m);
    apply2(re, im, 1 << (3 - k), 8, m);
  }
#pragma unroll
  for (int lq = 0; lq <= 1; ++lq)
#pragma unroll
    for (int tq = 2; tq <= 3; ++tq)
      cnot_c(re, im, 1 << (3 - lq), 1 << (3 - tq), 4);
  float z2 = 0.f, z3 = 0.f;
  for (int i = 0; i < 16; ++i) {
    float p = re[i]*re[i] + im[i]*im[i];
    z2 += (i & 2) ? -p : p;   // qubit2 -> bit1
    z3 += (i & 1) ? -p : p;   // qubit3 -> bit0
  }
  z[0] = z2; z[1] = z3;
}

// ============ kernel 1: build the fixed 128x128 real unitary of D's depth loop ============
__global__ __launch_bounds__(128) void k_build_unitary(const float* __restrict__ wD,
                                                       float* __restrict__ U) {
  __shared__ float u[128][128];
  int j = threadIdx.x;
  float* col = u[j];                       // column j = U e_j (Ry/CNOT are real)
  for (int i = 0; i < 128; ++i) col[i] = (i == j) ? 1.f : 0.f;
  for (int d = 0; d < 4; ++d) {
    for (int tq = 4; tq <= 6; ++tq)
      for (int lq = 0; lq <= 3; ++lq)
        cnot_r(col, 1 << (6 - tq), 1 << (6 - lq), 32);
    for (int k = 0; k < 7; ++k) {
      float c = __cosf(0.5f * wD[d*7 + k]);
      float s = __sinf(0.5f * wD[d*7 + k]);
      int mask = 1 << (6 - k);
      for (int p = 0; p < 64; ++p) {
        int i0 = ((p & ~(mask - 1)) << 1) | (p & (mask - 1));
        int i1 = i0 | mask;
        float r0 = col[i0], r1 = col[i1];
        col[i0] = c*r0 - s*r1;
        col[i1] = s*r0 + c*r1;
      }
    }
    for (int lq = 0; lq <= 3; ++lq)
      for (int tq = 4; tq <= 6; ++tq)
        cnot_r(col, 1 << (6 - lq), 1 << (6 - tq), 32);
  }
  for (int i = 0; i < 128; ++i) U[i*128 + j] = col[i];   // U[out][in], coalesced over j
}

// ============ kernel 2: per-element sim + WMMA application of U ============
__global__ __launch_bounds__(32) void k_qae_main(const float* __restrict__ x,
                                                 const float* __restrict__ wA,
                                                 const float* __restrict__ wB,
                                                 const float* __restrict__ wC,
                                                 const float* __restrict__ U,
                                                 float* __restrict__ out, int Btot) {
  __shared__ float sRe[32][130];           // 130 = pad -> conflict-free bank stride 2
  __shared__ float sIm[32][130];
  const int lane = threadIdx.x;
  const int base = blockIdx.x * 32;
  const int b = base + lane;
  const int bc = (b < Btot) ? b : (Btot - 1);
  const float* xrow = x + (long)bc * 56;
  __builtin_prefetch(U + (lane << 9), 0, 0);   // warm L2 with the 64KB unitary

  float* re = sRe[lane];
  float* im = sIm[lane];
  float z[2];
  run_block4(xrow, cA_PT, cA_ETA, cA_PHI, wA, re, im, z);
  if (b < Btot) { out[(long)b*9 + 0] = z[0]; out[(long)b*9 + 1] = z[1]; }
  run_block4(xrow, cB_PT, cB_ETA, cB_PHI, wB, re, im, z);
  if (b < Btot) { out[(long)b*9 + 2] = z[0]; out[(long)b*9 + 3] = z[1]; }
  run_block4(xrow, cC_PT, cC_ETA, cC_PHI, wC, re, im, z);
  if (b < Btot) { out[(long)b*9 + 4] = z[0]; out[(long)b*9 + 5] = z[1]; }

  // ---- block D data-dependent layers (7 qubits, 128 amps in LDS) ----
  for (int i = 0; i < 128; ++i) { re[i] = 0.f; im[i] = 0.f; }
  re[0] = 1.f;
  {
    C2 a[4], bm[4], t[4], m[4];
#pragma unroll
    for (int k = 0; k < 7; ++k) {
      int mask = 1 << (6 - k);
      float eta = (cD_ETA[k] >= 0) ? xrow[cD_ETA[k]] : 0.f;
      rot_rx(eta, a); rot_ry(xrow[cD_PT[k]], bm); mm2(bm, a, t);
      rot_rz(xrow[cD_PHI[k]], a); mm2(a, t, m);
      apply2(re, im, mask, 64, m);
    }
#pragma unroll
    for (int k = 0; k < 6; ++k)
      cnot_c(re, im, 1 << (6 - k), 1 << (5 - k), 32);
#pragma unroll
    for (int k = 0; k < 7; ++k) {
      int mask = 1 << (6 - k);
      float eta = (cD_ETA[k] >= 0) ? xrow[cD_ETA[k]] : 0.f;
      rot_rz(eta, a); rot_ry(xrow[cD_PT[k]], bm); mm2(bm, a, t);
      rot_rx(xrow[cD_PHI[k]], a); mm2(a, t, m);
      apply2(re, im, mask, 64, m);
    }
  }
  __syncthreads();

  // ---- apply U via V_WMMA_F32_16X16X4_F32:  out[b,n] = sum_k s[b,k] * U[n,k] ----
  // A 16x4 f32 layout: lanes 0-15 -> M=0..15 with K={0,1}; lanes 16-31 -> K={2,3}
  // B 4x16 f32 layout (assumed): VGPR r, lanes 0-15 K=2r, lanes 16-31 K=2r+1; N=lane&15
  // C/D 16x16 f32: lane n (0-15) VGPR r -> M=r; lanes 16-31 -> M=r+8
  const int half = lane >> 4;
  const int ln = lane & 15;
#pragma unroll
  for (int mt = 0; mt < 2; ++mt) {
    float acc0[8], acc1[8], acc2[8];
#pragma unroll
    for (int r = 0; r < 8; ++r) { acc0[r] = 0.f; acc1[r] = 0.f; acc2[r] = 0.f; }
    const float* aRrow = sRe[mt*16 + ln];
    const float* aIrow = sIm[mt*16 + ln];
    for (int nt = 0; nt < 8; ++nt) {
      v8f cr = {0.f,0.f,0.f,0.f,0.f,0.f,0.f,0.f};
      v8f ci = {0.f,0.f,0.f,0.f,0.f,0.f,0.f,0.f};
      const float* Urow = U + (nt*16 + ln) * 128;
#pragma unroll 4
      for (int kk = 0; kk < 32; ++kk) {
        const int kb = kk * 4;
        v2f aR, aI, bv;
        aR.x = aRrow[kb + half*2 + 0];
        aR.y = aRrow[kb + half*2 + 1];
        aI.x = aIrow[kb + half*2 + 0];
        aI.y = aIrow[kb + half*2 + 1];
        bv.x = Urow[kb + half + 0];
        bv.y = Urow[kb + half + 2];
#ifdef HAVE_WMMA_F32X4
        cr = __builtin_amdgcn_wmma_f32_16x16x4_f32(false, aR, false, bv,
                                                   (short)0, cr, false, false);
        ci = __builtin_amdgcn_wmma_f32_16x16x4_f32(false, aI, false, bv,
                                                   (short)0, ci, false, false);
#else
        // scalar fallback (keeps compile green if the builtin probe fails)
#pragma unroll
        for (int r = 0; r < 8; ++r) {
          cr[r] += aR.x * bv.x + aR.y * bv.y;
          ci[r] += aI.x * bv.x + aI.y * bv.y;
        }
#endif
      }
      // amp index n is uniform per lane -> Z signs are lane-uniform
      int ng = nt*16 + ln;
      float s0 = (ng & 4) ? -1.f : 1.f;   // qubit 4 -> bit 2
      float s1 = (ng & 2) ? -1.f : 1.f;   // qubit 5 -> bit 1
      float s2 = (ng & 1) ? -1.f : 1.f;   // qubit 6 -> bit 0
#pragma unroll
      for (int r = 0; r < 8; ++r) {
        float p = cr[r]*cr[r] + ci[r]*ci[r];
        acc0[r] += s0 * p; acc1[r] += s1 * p; acc2[r] += s2 * p;
      }
    }
    // reduce over the 16 lanes of each half (rows m=r in lanes 0-15, m=r+8 in 16-31)
#pragma unroll
    for (int sh = 1; sh < 16; sh <<= 1) {
#pragma unroll
      for (int r = 0; r < 8; ++r) {
        acc0[r] += __shfl_xor(acc0[r], sh, 32);
        acc1[r] += __shfl_xor(acc1[r], sh, 32);
        acc2[r] += __shfl_xor(acc2[r], sh, 32);
      }
    }
    if (ln == 0) {
#pragma unroll
      for (int r = 0; r < 8; ++r) {
        long e = (long)base + mt*16 + half*8 + r;
        if (e < Btot) {
          out[e*9 + 6] = acc0[r];
          out[e*9 + 7] = acc1[r];
          out[e*9 + 8] = acc2[r];
        }
      }
    }
  }
}

extern "C" void kernel_launch(void* const* d_in, const int* in_sizes, int n_in,
                              void* d_out, int out_size, void* d_ws, size_t ws_size,
                              hipStream_t stream) {
  const float* x  = (const float*)d_in[0];
  const float* wA = (const float*)d_in[1];
  const float* wB = (const float*)d_in[2];
  const float* wC = (const float*)d_in[3];
  const float* wD = (const float*)d_in[4];
  float* out = (float*)d_out;
  float* U = (float*)d_ws;                 // 128*128 f32 = 64 KB scratch
  int B = in_sizes[0] / 56;
  k_build_unitary<<<1, 128, 0, stream>>>(wD, U);
  int nblk = (B + 31) / 32;
  k_qae_main<<<nblk, 32, 0, stream>>>(x, wA, wB, wC, U, out, B);
}